// TimeAwareConv1d_47167330845495
// MI455X (gfx1250) — compile-verified
//
#include <hip/hip_runtime.h>

typedef __attribute__((ext_vector_type(2))) float v2f;
typedef __attribute__((ext_vector_type(8))) float v8f;

#define BB 16
#define II 64
#define CC 4
#define LL 4096
#define OO 64
#define KW 3
#define NBLK 128
#define XSTR 136  // LDS row stride (floats); 2*136 % 64 == 16 -> half-wave bank ranges disjoint

// Pre-transpose weights: wT[((c*3 + k)*64 + i)*64 + o] = w[((o*64 + i)*4 + c)*3 + k]
__global__ void prep_weights_kernel(const float* __restrict__ w, float* __restrict__ wT) {
  int idx = blockIdx.x * blockDim.x + threadIdx.x;
  if (idx >= OO * II * CC * KW) return;
  int o = idx & 63;
  int i = (idx >> 6) & 63;
  int k = (idx >> 12) % KW;
  int c = idx / (OO * II * KW);
  wT[idx] = w[((size_t)(o * II + i) * CC + c) * KW + k];
}

__global__ __launch_bounds__(256)
void tconv_wmma_kernel(const float* __restrict__ x, const float* __restrict__ tm,
                       const float* __restrict__ wT, float* __restrict__ out) {
  __shared__ float sX[II][XSTR];  // 64 x 136 x 4B = 34.8 KB

  const int tid   = threadIdx.x;
  const int lane  = tid & 31;
  const int wave  = tid >> 5;
  const int m0    = (wave & 3) * 16;  // o-tile base
  const int nhalf = wave >> 2;        // 0 or 1 -> 64-col half of the 128-col block
  const int nb    = blockIdx.x;       // L tile: 0..31
  const int c     = blockIdx.y;       // 0..3
  const int b     = blockIdx.z;       // 0..15
  const int l0    = nb * NBLK;

  // ---- Stage x rows [i][l0-1 .. l0+NBLK] into LDS (halo + zero pad at edges) ----
  const float* xb = x + ((size_t)b * II * CC + c) * LL;  // row i lives at xb + i*CC*LL
  for (int idx = tid; idx < II * (NBLK + 2); idx += 256) {
    int i = idx / (NBLK + 2);
    int n = idx - i * (NBLK + 2);
    int l = l0 - 1 + n;
    float v = 0.0f;
    if (l >= 0 && l < LL) v = xb[(size_t)i * (CC * LL) + l];
    sX[i][n] = v;
  }
  __syncthreads();

  // ---- GEMM: out[o, l] = sum_k sum_i wT[c][k][i][o] * sX[i][l - l0 + k] ----
  v8f acc[4] = {};
  const int am   = m0 + (lane & 15);   // o index this lane holds in A frags
  const int aoff = (lane >> 4) * 2;    // K-slot base: lanes 0-15 -> slots 0,1; 16-31 -> 2,3
  const int bn   = lane & 15;          // n index this lane holds in B frags
  const float* wc = wT + (size_t)c * (KW * II * OO);

  for (int k = 0; k < KW; ++k) {
    const float* wk = wc + (size_t)k * (II * OO);
    // Preload all A fragments for this k (16 steps over i, 4 i's per step)
    v2f afr[16];
#pragma unroll
    for (int f = 0; f < 16; ++f) {
      int i0 = 4 * f + aoff;
      afr[f].x = wk[(size_t)i0 * OO + am];
      afr[f].y = wk[(size_t)(i0 + 1) * OO + am];
    }
#pragma unroll
    for (int f = 0; f < 16; ++f) {
      int i0  = 4 * f + aoff;
      int col = nhalf * 64 + bn + k;  // sX col for this lane's n, shifted by k
      v2f b0, b1, b2, b3;
      b0.x = sX[i0][col     ]; b0.y = sX[i0 + 1][col     ];
      b1.x = sX[i0][col + 16]; b1.y = sX[i0 + 1][col + 16];
      b2.x = sX[i0][col + 32]; b2.y = sX[i0 + 1][col + 32];
      b3.x = sX[i0][col + 48]; b3.y = sX[i0 + 1][col + 48];
      acc[0] = __builtin_amdgcn_wmma_f32_16x16x4_f32(false, afr[f], false, b0, (short)0, acc[0], false, false);
      acc[1] = __builtin_amdgcn_wmma_f32_16x16x4_f32(false, afr[f], false, b1, (short)0, acc[1], false, false);
      acc[2] = __builtin_amdgcn_wmma_f32_16x16x4_f32(false, afr[f], false, b2, (short)0, acc[2], false, false);
      acc[3] = __builtin_amdgcn_wmma_f32_16x16x4_f32(false, afr[f], false, b3, (short)0, acc[3], false, false);
    }
  }

  // ---- Epilogue: scale by time[b,c,l], store (b,o,c,l) ----
  // C/D layout: VGPR r holds M = r + 8*(lane>>4), N = lane&15
  const float* tp = tm + ((size_t)b * CC + c) * LL + l0;
  const int rowadd = (lane >> 4) * 8;
#pragma unroll
  for (int t = 0; t < 4; ++t) {
    int col = nhalf * 64 + t * 16 + bn;
    float tv = tp[col];
    size_t obase = (((size_t)b * OO + m0 + rowadd) * CC + c) * LL + l0 + col;
#pragma unroll
    for (int r = 0; r < 8; ++r) {
      out[obase + (size_t)r * (CC * LL)] = acc[t][r] * tv;
    }
  }
}

extern "C" void kernel_launch(void* const* d_in, const int* in_sizes, int n_in,
                              void* d_out, int out_size, void* d_ws, size_t ws_size,
                              hipStream_t stream) {
  (void)in_sizes; (void)n_in; (void)out_size; (void)ws_size;
  const float* x  = (const float*)d_in[0];  // (16,64,4,4096)
  const float* tm = (const float*)d_in[1];  // (16,4,4096)
  const float* w  = (const float*)d_in[2];  // (64,64,4,3)
  float* out = (float*)d_out;               // (16,64,4,4096)
  float* wT  = (float*)d_ws;                // 49152 floats = 192 KB scratch

  int wn = OO * II * CC * KW;
  prep_weights_kernel<<<dim3((wn + 255) / 256), 256, 0, stream>>>(w, wT);
  tconv_wmma_kernel<<<dim3(LL / NBLK, CC, BB), 256, 0, stream>>>(x, tm, wT, out);
}